// QuantizationLayer_28063316312405
// MI455X (gfx1250) — compile-verified
//
#include <hip/hip_runtime.h>

// Event-camera quantization layer for MI455X (gfx1250).
// Pipeline: zero-init -> per-batch tmax (LDS ds_max + global atomicMax on
// float bits) -> scatter (double-buffered global_load_async_to_lds_b128
// staging; per event: 1x global_atomic_add_u32 into full-res histogram +
// 1x global_atomic_add_u64 packed {count:24 | fixed-point sum(t):40} into
// half-res voxels) -> finalize (unpack + 2x2 signed pooling).

#define EV_W   240
#define EV_H   180
#define EV_B   16
#define NPIX   (EV_B * EV_H * EV_W)               // 691200
#define NVOX   (EV_B * (EV_H / 2) * (EV_W / 2))   // 172800
#define CHUNK  1024                               // events staged per chunk
#define TPB    256
#define EPB    8192                               // events per block

#define T_SCALE   16777216.0f                     // 2^24 fixed-point for t
#define T_INVSCL  (1.0f / 16777216.0f)
#define T_MASK    0xFFFFFFFFFFull                 // low 40 bits
#define CNT_ONE   (1ull << 40)

// ---------------------------------------------------------------- zero init
__global__ void ev_zero(unsigned* __restrict__ p, int nwords) {
  int i = blockIdx.x * blockDim.x + threadIdx.x;
  if (i < nwords) p[i] = 0u;
}

// ---------------------------------------------------------------- pass 1: tmax
// b is sorted, so each thread's contiguous run of events touches 1-2 batches:
// keep a register running max, flush to LDS (ds_max_u32) on batch change,
// then <=16 global atomicMax per block. Positive floats compare as uints.
// Side effect: streams the 160 MB event array into the 192 MB L2 for pass 2.
__global__ __launch_bounds__(TPB) void ev_tmax(const float* __restrict__ ev,
                                               int n,
                                               unsigned* __restrict__ tmaxbits) {
  __shared__ unsigned sm[EV_B];
  int tid = threadIdx.x;
  if (tid < EV_B) sm[tid] = 0u;
  __syncthreads();

  long long s = (long long)blockIdx.x * EPB;
  long long e = s + EPB;
  if (e > n) e = n;
  long long ts = s + (long long)tid * (EPB / TPB);
  long long te = ts + (EPB / TPB);
  if (te > e) te = e;

  int curb = -1;
  unsigned curm = 0u;
  for (long long i = ts; i < te; ++i) {
    float t = ev[i * 5 + 2];
    int bi = (int)ev[i * 5 + 4];
    unsigned tb = __float_as_uint(t);
    if (bi != curb) {
      if (curb >= 0) atomicMax(&sm[curb], curm);
      curb = bi;
      curm = tb;
    } else if (tb > curm) {
      curm = tb;
    }
  }
  if (curb >= 0) atomicMax(&sm[curb], curm);
  __syncthreads();
  if (tid < EV_B) {
    unsigned v = sm[tid];
    if (v) atomicMax(&tmaxbits[tid], v);
  }
}

// ------------------------------------------------------- async chunk staging
// Stage CHUNK events (CHUNK*20 bytes) into LDS as 5*TPB lane-granular 16B
// async copies. ASYNCcnt-tracked; completion enforced with s_wait_asynccnt.
__device__ __forceinline__ void issue_async(const float* __restrict__ gsrc,
                                            float* ldst, int tid) {
  unsigned lbase = (unsigned)(size_t)ldst;  // low 32 bits of flat addr == LDS offset
  unsigned long long gbase = (unsigned long long)(size_t)gsrc;
#pragma unroll
  for (int j = 0; j < 5; ++j) {
    unsigned loff = lbase + (unsigned)((j * TPB + tid) * 16);
    unsigned long long ga = gbase + (unsigned long long)((j * TPB + tid) * 16);
    asm volatile("global_load_async_to_lds_b128 %0, %1, off"
                 :: "v"(loff), "v"(ga) : "memory");
  }
}

// ----------------------------------------------------------- per-event update
__device__ __forceinline__ void ev_accumulate(float x, float y, float t, float b,
                                              const float* rt,
                                              unsigned* __restrict__ cont,
                                              unsigned long long* __restrict__ pk) {
  int xi = (int)x, yi = (int)y, bi = (int)b;
  int idx_c = xi + EV_W * yi + (EV_W * EV_H) * bi;       // x + 240y + 43200b
  int idx_k = (xi >> 1) + 60 * yi + 10800 * bi;          // x//2 + 240y//4 + 43200b//4
  float tn = t * rt[bi];                                 // in [0,1]
  unsigned long long packed = CNT_ONE | (unsigned long long)(unsigned)(tn * T_SCALE);
  atomicAdd(&cont[idx_c], 1u);                           // global_atomic_add_u32
  atomicAdd(&pk[idx_k], packed);                         // global_atomic_add_u64
}

// -------------------------------------------------------------- pass 2: scatter
__global__ __launch_bounds__(TPB) void ev_scatter(const float* __restrict__ ev,
                                                  int n,
                                                  const unsigned* __restrict__ tmaxbits,
                                                  unsigned* __restrict__ cont,
                                                  unsigned long long* __restrict__ pk) {
  __shared__ float buf[2][CHUNK * 5];   // 40 KB double buffer
  __shared__ float rt[EV_B];
  int tid = threadIdx.x;
  if (tid < EV_B) {
    unsigned bbits = tmaxbits[tid];
    rt[tid] = bbits ? (1.0f / __uint_as_float(bbits)) : 0.0f;
  }

  long long s = (long long)blockIdx.x * EPB;
  long long e = s + EPB;
  if (e > n) e = n;
  int total = (int)(e > s ? e - s : 0);
  int nc = total / CHUNK;

  if (nc > 0) issue_async(ev + s * 5, &buf[0][0], tid);

  for (int c = 0; c < nc; ++c) {
    if (c + 1 < nc) {
      issue_async(ev + (s + (long long)(c + 1) * CHUNK) * 5, &buf[(c + 1) & 1][0], tid);
      asm volatile("s_wait_asynccnt 5" ::: "memory");  // chunk c done, c+1 in flight
    } else {
      asm volatile("s_wait_asynccnt 0" ::: "memory");
    }
    __syncthreads();  // all waves waited -> chunk c fully resident in LDS

    // 4 events per thread: 80 B at 16B alignment -> 5x ds_load_b128
    const float* p = &buf[c & 1][tid * 20];
    float4 v0 = ((const float4*)p)[0];
    float4 v1 = ((const float4*)p)[1];
    float4 v2 = ((const float4*)p)[2];
    float4 v3 = ((const float4*)p)[3];
    float4 v4 = ((const float4*)p)[4];
    float f[20] = {v0.x, v0.y, v0.z, v0.w, v1.x, v1.y, v1.z, v1.w,
                   v2.x, v2.y, v2.z, v2.w, v3.x, v3.y, v3.z, v3.w,
                   v4.x, v4.y, v4.z, v4.w};
#pragma unroll
    for (int j = 0; j < 4; ++j)
      ev_accumulate(f[5 * j], f[5 * j + 1], f[5 * j + 2], f[5 * j + 4],
                    rt, cont, pk);
    __syncthreads();  // done reading buf[c&1] before chunk c+2 overwrites it
  }

  __syncthreads();  // rt visibility for the (rare) nc==0 remainder path
  for (int i = nc * CHUNK + tid; i < total; i += TPB) {
    const float* q = ev + (s + i) * 5;
    ev_accumulate(q[0], q[1], q[2], q[4], rt, cont, pk);
  }
}

// ------------------------------------------------------------- pass 3: finalize
// One thread per half-res voxel: unpacks {count|sum(t)}, emits counter/timer/
// diff_x/diff_y, and copies its private 2x2 container block to float output
// (each container pixel written exactly once).
__global__ void ev_finalize(const unsigned* __restrict__ cont,
                            const unsigned long long* __restrict__ pk,
                            float* __restrict__ out, int nvox) {
  int v = blockIdx.x * blockDim.x + threadIdx.x;
  if (v >= nvox) return;
  int j = v % (EV_W / 2);
  int r = v / (EV_W / 2);
  int i = r % (EV_H / 2);
  int b = r / (EV_H / 2);

  int base = b * (EV_H * EV_W) + (2 * i) * EV_W + 2 * j;
  int c00 = (int)cont[base], c01 = (int)cont[base + 1];
  int c10 = (int)cont[base + EV_W], c11 = (int)cont[base + EV_W + 1];

  float* ocont = out;                    // [B,H,W]      691200
  float* ocntr = out + NPIX;             // [B,H/2,W/2]  172800
  float* otim  = ocntr + NVOX;
  float* odx   = otim + NVOX;
  float* ody   = odx + NVOX;

  ocont[base]            = (float)c00;
  ocont[base + 1]        = (float)c01;
  ocont[base + EV_W]     = (float)c10;
  ocont[base + EV_W + 1] = (float)c11;

  unsigned long long pv = pk[v];
  unsigned cnt = (unsigned)(pv >> 40);
  float tsum = (float)(pv & T_MASK) * T_INVSCL;
  ocntr[v] = (float)cnt;
  otim[v]  = tsum / (cnt == 0u ? 1.0f : (float)cnt);
  odx[v]   = (float)((c10 - c11) + (c00 - c01));
  ody[v]   = (float)((c00 - c10) + (c01 - c11));
}

// --------------------------------------------------------------------- launch
extern "C" void kernel_launch(void* const* d_in, const int* in_sizes, int n_in,
                              void* d_out, int out_size, void* d_ws, size_t ws_size,
                              hipStream_t stream) {
  const float* ev = (const float*)d_in[0];
  int n = in_sizes[0] / 5;

  unsigned* tmaxbits = (unsigned*)d_ws;                       // 16 words
  unsigned* cont = (unsigned*)d_ws + 16;                      // NPIX words
  unsigned long long* pk =                                    // NVOX u64 (8B aligned)
      (unsigned long long*)(cont + NPIX);
  float* out = (float*)d_out;

  int init_words = 16 + NPIX + 2 * NVOX;                      // ~4.15 MB of scratch
  ev_zero<<<(init_words + 255) / 256, 256, 0, stream>>>((unsigned*)d_ws, init_words);

  int nb = (n + EPB - 1) / EPB;
  ev_tmax<<<nb, TPB, 0, stream>>>(ev, n, tmaxbits);
  ev_scatter<<<nb, TPB, 0, stream>>>(ev, n, tmaxbits, cont, pk);
  ev_finalize<<<(NVOX + 255) / 256, 256, 0, stream>>>(cont, pk, out, NVOX);
}